// TDOPConv_37838661878277
// MI455X (gfx1250) — compile-verified
//
#include <hip/hip_runtime.h>
#include <math.h>

// ---------------------------------------------------------------------------
// TDOPConv for MI455X (gfx1250, wave32).  All GEMMs use V_WMMA_F32_16X16X4_F32
// (fp32 matrix cores, exact precision).  Graph scatter uses f32 global atomics;
// the 51.2MB feature buffers fit in the 192MB L2 so the random gather/scatter
// runs out of L2, not HBM.
// ---------------------------------------------------------------------------

typedef __attribute__((ext_vector_type(2))) float v2f;
typedef __attribute__((ext_vector_type(8))) float v8f;

#define DD 128
#define THETA0 0.5f
#define THETA1 0.3f
#define THETA2 0.2f
#define NEG_SLOPE 0.01f

__device__ __forceinline__ v8f wmma4(v2f a, v2f b, v8f c) {
  // D = A(16x4,f32) * B(4x16,f32) + C(16x16,f32)
  return __builtin_amdgcn_wmma_f32_16x16x4_f32(
      /*neg_a=*/false, a, /*neg_b=*/false, b,
      /*c_mod=*/(short)0, c, /*reuse_a=*/false, /*reuse_b=*/false);
}

// ---------------------------------------------------------------- utilities
__global__ void zero_f32(float* __restrict__ p, int n) {
  int i = blockIdx.x * blockDim.x + threadIdx.x;
  if (i < n) p[i] = 0.0f;
}

__global__ void deg_kernel(const int* __restrict__ dst, float* __restrict__ degs, int E) {
  int i = blockIdx.x * blockDim.x + threadIdx.x;
  if (i < E) atomicAdd(&degs[dst[i]], 1.0f);
}

__global__ void dinv_kernel(float* __restrict__ degs, int n) {
  int i = blockIdx.x * blockDim.x + threadIdx.x;
  if (i < n) {
    float d = degs[i];
    d = d < 1.0f ? 1.0f : d;
    degs[i] = rsqrtf(d);   // in-place: degs becomes dinv
  }
}

// ------------------------------------------------- fused MLP + xk seed (WMMA)
// One wave per 16-row tile.  H = relu(X*W1^T + b1) kept in LDS, 128->2 head,
// softmax prob, xk = x + alpha*((1-p)*emb0 + p*emb1), hi seeded with theta0*xk.
__global__ void __launch_bounds__(128)
mlp_xk_kernel(const float* __restrict__ x,  const float* __restrict__ w1,
              const float* __restrict__ b1, const float* __restrict__ w2,
              const float* __restrict__ b2, const float* __restrict__ emb,
              const float* __restrict__ alpha,
              float* __restrict__ xk, float* __restrict__ hi, int ntiles) {
  __shared__ float lds_h[4][16 * DD];
  __shared__ float lds_lg[4][32];
  __shared__ float lds_p[4][16];

  const int wave = threadIdx.x >> 5;
  const int lane = threadIdx.x & 31;
  const int tile = blockIdx.x * 4 + wave;
  if (tile >= ntiles) return;            // wave-uniform: EXEC stays all-ones

  const int row0 = tile * 16;
  const int mA   = lane & 15;            // row (A) / col (B,C) within tile
  const int koff = (lane >> 4) * 2;      // half-wave k offset (ISA 16x4 layout)
  const float* arow = x + (size_t)(row0 + mA) * DD;

  v8f acc[8];
#pragma unroll
  for (int j = 0; j < 8; ++j) acc[j] = (v8f){0,0,0,0,0,0,0,0};

  for (int k = 0; k < DD; k += 4) {
    v2f a; a.x = arow[k + koff]; a.y = arow[k + koff + 1];
#pragma unroll
    for (int j = 0; j < 8; ++j) {
      // B(kk,n) = W1^T[kk][n] = w1[n*DD + kk], n = 16j + mA
      const float* bp = w1 + (size_t)(16 * j + mA) * DD + (k + koff);
      v2f b; b.x = bp[0]; b.y = bp[1];
      acc[j] = wmma4(a, b, acc[j]);
    }
  }

  // bias + relu -> LDS (C layout: vgpr r holds M = r + 8*(lane>=16), N = mA)
  const int rbase = (lane >> 4) * 8;
#pragma unroll
  for (int j = 0; j < 8; ++j) {
    const int col  = 16 * j + mA;
    const float bb = b1[col];
#pragma unroll
    for (int r = 0; r < 8; ++r) {
      float h = acc[j][r] + bb;
      lds_h[wave][(rbase + r) * DD + col] = h > 0.0f ? h : 0.0f;
    }
  }

  // logits: lane = (cls<<4)|row   (same-wave LDS ops are in-order on CDNA5)
  {
    const int rowL = lane & 15;
    const int cls  = lane >> 4;
    const float* w2r = w2 + cls * DD;
    float s = 0.0f;
    for (int c = 0; c < DD; ++c) s += lds_h[wave][rowL * DD + c] * w2r[c];
    lds_lg[wave][lane] = s + b2[cls];
  }
  if (lane < 16) {
    float l0 = lds_lg[wave][lane];
    float l1 = lds_lg[wave][16 + lane];
    lds_p[wave][lane] = 1.0f / (1.0f + expf(l0 - l1));  // softmax[:,1]
  }

  const float al = alpha[0];
  for (int idx = lane; idx < 16 * DD; idx += 32) {
    const int r = idx >> 7, c = idx & (DD - 1);
    const float p = lds_p[wave][r];
    const size_t o = (size_t)(row0 + r) * DD + c;
    const float v = x[o] + al * ((1.0f - p) * emb[c] + p * emb[DD + c]);
    xk[o] = v;
    hi[o] = THETA0 * v;
  }
}

// -------------------------------------------------- Laplacian scatter/update
// one wave per edge; float4 gather, 4x global_atomic_add_f32 scatter
__global__ void __launch_bounds__(256)
scatter_kernel(const int* __restrict__ src, const int* __restrict__ dst,
               const float* __restrict__ xk, const float* __restrict__ dinv,
               float* __restrict__ msg, int E) {
  int t = blockIdx.x * blockDim.x + threadIdx.x;
  int edge = t >> 5;
  if (edge >= E) return;
  int lane = t & 31;
  int s = src[edge], d = dst[edge];
  float sc = dinv[s];
  const float4* xr = (const float4*)(xk + (size_t)s * DD);
  float4 v = xr[lane];
  float* mr = msg + (size_t)d * DD + lane * 4;
  atomicAdd(mr + 0, v.x * sc);
  atomicAdd(mr + 1, v.y * sc);
  atomicAdd(mr + 2, v.z * sc);
  atomicAdd(mr + 3, v.w * sc);
}

__global__ void update_kernel(float* __restrict__ xk, const float* __restrict__ msg,
                              const float* __restrict__ dinv, float* __restrict__ hi,
                              float theta, int total) {
  int i = blockIdx.x * blockDim.x + threadIdx.x;
  if (i < total) {
    int row = i >> 7;
    float v = xk[i] - msg[i] * dinv[row];
    xk[i] = v;
    hi[i] += theta * v;
  }
}

// ------------------------------------- GEMM2: outputs = [hi | x] @ weights
__global__ void __launch_bounds__(256)
gemm2_kernel(const float* __restrict__ hi, const float* __restrict__ x,
             const float* __restrict__ weights, float* __restrict__ outputs,
             int ntiles) {
  const int wave = threadIdx.x >> 5;
  const int lane = threadIdx.x & 31;
  const int tile = blockIdx.x * 8 + wave;
  if (tile >= ntiles) return;
  const int row0 = tile * 16;
  const int mA = lane & 15, koff = (lane >> 4) * 2;

  v8f acc[8];
#pragma unroll
  for (int j = 0; j < 8; ++j) acc[j] = (v8f){0,0,0,0,0,0,0,0};

  const float* a1 = hi + (size_t)(row0 + mA) * DD;
  for (int k = 0; k < DD; k += 4) {
    v2f a; a.x = a1[k + koff]; a.y = a1[k + koff + 1];
#pragma unroll
    for (int j = 0; j < 8; ++j) {
      // B(kk,n) = weights[kk*DD + n]
      const float* bp = weights + (size_t)(k + koff) * DD + 16 * j + mA;
      v2f b; b.x = bp[0]; b.y = bp[DD];
      acc[j] = wmma4(a, b, acc[j]);
    }
  }
  const float* a2 = x + (size_t)(row0 + mA) * DD;
  for (int k = 0; k < DD; k += 4) {
    v2f a; a.x = a2[k + koff]; a.y = a2[k + koff + 1];
#pragma unroll
    for (int j = 0; j < 8; ++j) {
      const float* bp = weights + (size_t)(DD + k + koff) * DD + 16 * j + mA;
      v2f b; b.x = bp[0]; b.y = bp[DD];
      acc[j] = wmma4(a, b, acc[j]);
    }
  }
  const int rbase = (lane >> 4) * 8;
#pragma unroll
  for (int j = 0; j < 8; ++j)
#pragma unroll
    for (int r = 0; r < 8; ++r)
      outputs[(size_t)(row0 + rbase + r) * DD + 16 * j + mA] = acc[j][r];
}

// --- GEMM3: out = leaky_relu([e | outputs] @ lin_w^T + lin_b) + x0 -> d_out
__global__ void __launch_bounds__(256)
gemm3_kernel(const float* __restrict__ e, const float* __restrict__ outputs,
             const float* __restrict__ lin_w, const float* __restrict__ lin_b,
             const float* __restrict__ x0, float* __restrict__ out, int ntiles) {
  const int wave = threadIdx.x >> 5;
  const int lane = threadIdx.x & 31;
  const int tile = blockIdx.x * 8 + wave;
  if (tile >= ntiles) return;
  const int row0 = tile * 16;
  const int mA = lane & 15, koff = (lane >> 4) * 2;

  v8f acc[8];
#pragma unroll
  for (int j = 0; j < 8; ++j) acc[j] = (v8f){0,0,0,0,0,0,0,0};

  const float* a1 = e + (size_t)(row0 + mA) * DD;
  for (int k = 0; k < DD; k += 4) {
    v2f a; a.x = a1[k + koff]; a.y = a1[k + koff + 1];
#pragma unroll
    for (int j = 0; j < 8; ++j) {
      // B(kk,n) = lin_w[n*2DD + kk]
      const float* bp = lin_w + (size_t)(16 * j + mA) * (2 * DD) + (k + koff);
      v2f b; b.x = bp[0]; b.y = bp[1];
      acc[j] = wmma4(a, b, acc[j]);
    }
  }
  const float* a2 = outputs + (size_t)(row0 + mA) * DD;
  for (int k = 0; k < DD; k += 4) {
    v2f a; a.x = a2[k + koff]; a.y = a2[k + koff + 1];
#pragma unroll
    for (int j = 0; j < 8; ++j) {
      const float* bp = lin_w + (size_t)(16 * j + mA) * (2 * DD) + (DD + k + koff);
      v2f b; b.x = bp[0]; b.y = bp[1];
      acc[j] = wmma4(a, b, acc[j]);
    }
  }
  const int rbase = (lane >> 4) * 8;
#pragma unroll
  for (int j = 0; j < 8; ++j) {
    const int col = 16 * j + mA;
    const float bb = lin_b[col];
#pragma unroll
    for (int r = 0; r < 8; ++r) {
      const size_t o = (size_t)(row0 + rbase + r) * DD + col;
      float v = acc[j][r] + bb;
      v = v > 0.0f ? v : NEG_SLOPE * v;
      out[o] = v + x0[o];
    }
  }
}

// ---------------------------------------------------------------------------
extern "C" void kernel_launch(void* const* d_in, const int* in_sizes, int n_in,
                              void* d_out, int out_size, void* d_ws, size_t ws_size,
                              hipStream_t stream) {
  const int*   src     = (const int*)  d_in[0];
  const int*   dst     = (const int*)  d_in[1];
  const float* x0      = (const float*)d_in[2];
  const float* x       = (const float*)d_in[3];
  const float* e       = (const float*)d_in[4];
  // d_in[5] labels, d_in[6] nid: unused by the reference
  const float* alpha   = (const float*)d_in[7];
  const float* emb     = (const float*)d_in[8];
  const float* w1      = (const float*)d_in[9];
  const float* b1      = (const float*)d_in[10];
  const float* w2      = (const float*)d_in[11];
  const float* b2      = (const float*)d_in[12];
  const float* weights = (const float*)d_in[13];
  const float* lin_w   = (const float*)d_in[14];
  const float* lin_b   = (const float*)d_in[15];

  const int E  = in_sizes[0];
  const int N  = in_sizes[2] / DD;     // 100000 (multiple of 16)
  const int ND = N * DD;
  const int ntiles = N / 16;

  float* ws   = (float*)d_ws;
  float* xk   = ws;                      // N*D
  float* msg  = ws + (size_t)ND;         // N*D, reused as `outputs` after passes
  float* dinv = ws + (size_t)2 * ND;     // N   (degs -> dinv in place)
  float* hi   = (float*)d_out;           // scratch; fully rewritten by gemm3
  float* outp = msg;

  const int T = 256;
  // degrees -> dinv
  zero_f32   <<<(N  + T - 1) / T, T, 0, stream>>>(dinv, N);
  deg_kernel <<<(E  + T - 1) / T, T, 0, stream>>>(dst, dinv, E);
  dinv_kernel<<<(N  + T - 1) / T, T, 0, stream>>>(dinv, N);
  // MLP head + xk, hi = theta0*xk
  mlp_xk_kernel<<<(ntiles + 3) / 4, 128, 0, stream>>>(x, w1, b1, w2, b2, emb,
                                                      alpha, xk, hi, ntiles);
  // Laplacian pass 1
  zero_f32<<<(ND + T - 1) / T, T, 0, stream>>>(msg, ND);
  scatter_kernel<<<(int)(((size_t)E * 32 + T - 1) / T), T, 0, stream>>>(src, dst, xk, dinv, msg, E);
  update_kernel<<<(ND + T - 1) / T, T, 0, stream>>>(xk, msg, dinv, hi, THETA1, ND);
  // Laplacian pass 2
  zero_f32<<<(ND + T - 1) / T, T, 0, stream>>>(msg, ND);
  scatter_kernel<<<(int)(((size_t)E * 32 + T - 1) / T), T, 0, stream>>>(src, dst, xk, dinv, msg, E);
  update_kernel<<<(ND + T - 1) / T, T, 0, stream>>>(xk, msg, dinv, hi, THETA2, ND);
  // outputs = [hi|x] @ weights ; out = leaky([e|outputs] @ lin_w^T + b) + x0
  gemm2_kernel<<<(ntiles + 7) / 8, 256, 0, stream>>>(hi, x, weights, outp, ntiles);
  gemm3_kernel<<<(ntiles + 7) / 8, 256, 0, stream>>>(e, outp, lin_w, lin_b, x0,
                                                     (float*)d_out, ntiles);
}